// LightSpatialAttention_64570538328398
// MI455X (gfx1250) — compile-verified
//
#include <hip/hip_runtime.h>
#include <math.h>

typedef __attribute__((ext_vector_type(2))) float v2f;
typedef __attribute__((ext_vector_type(8))) float v8f;

#define BATCH 8
#define CH    256
#define HH    160
#define WW    160
#define HWSZ  (HH * WW)          // 25600 pixels per plane
#define GROUPS_PER_B (HWSZ / 16) // 1600 16-pixel groups per batch

// ---------------------------------------------------------------------------
// Kernel 1: channel mean (via V_WMMA_F32_16X16X4_F32) + channel max.
// One wave32 handles 16 consecutive pixels for one batch, looping over the
// 256 channels 4 at a time.  B-matrix feed: pixel p's 4 channel values sit at
// lanes p and p+16 in the two B VGPRs, so column n == pixel n.  A is uniform
// 1/256, making D[m,n] = mean over channels of pixel n for every row m
// (invariant to A/B K-ordering details).
// Writes desc as [B, 2, H, W]: plane 0 = avg, plane 1 = max.
// ---------------------------------------------------------------------------
__global__ __launch_bounds__(256) void k_reduce(const float* __restrict__ x,
                                                float* __restrict__ desc) {
  const int wid  = threadIdx.x >> 5;
  const int lane = threadIdx.x & 31;
  const int gid  = blockIdx.x * 8 + wid;           // global wave id
  const int b    = gid / GROUPS_PER_B;
  const int base = (gid - b * GROUPS_PER_B) * 16;  // first pixel of this group
  const int p    = lane & 15;
  const int half = lane >> 4;
  const int hw   = base + p;

  const float* xb = x + (size_t)b * CH * HWSZ + hw;

  const float scale = 1.0f / 256.0f;
  v2f a; a[0] = scale; a[1] = scale;               // uniform A matrix
  v8f acc = {};                                    // f32 accumulator (C/D)
  float vmax = -3.402823466e+38f;

  for (int c = 0; c < CH; c += 4) {
    const int ch = c + 2 * half;                   // half 0: c,c+1  half 1: c+2,c+3
    float b0 = xb[(size_t)ch * HWSZ];
    float b1 = xb[(size_t)(ch + 1) * HWSZ];
    v2f bb; bb[0] = b0; bb[1] = b1;
    acc = __builtin_amdgcn_wmma_f32_16x16x4_f32(false, a, false, bb,
                                                (short)0, acc, false, false);
    vmax = fmaxf(vmax, fmaxf(b0, b1));
  }

  // combine the two wave halves' channel-max partials
  vmax = fmaxf(vmax, __shfl_xor(vmax, 16, 32));

  if (half == 0) {
    // acc[0] = row M=0 of D for column p == mean of pixel hw
    desc[(size_t)(b * 2 + 0) * HWSZ + hw] = acc[0];
    desc[(size_t)(b * 2 + 1) * HWSZ + hw] = vmax;
  }
}

// ---------------------------------------------------------------------------
// Kernel 2: fused conv7x7 + conv3x3 + sigmoid weight, then out = x * w.
// One block = one batch x 2-row strip (full width -> contiguous 320-float
// tile per channel -> fully coalesced float4 streaming).
// ---------------------------------------------------------------------------
__global__ __launch_bounds__(256) void k_conv_mul(const float* __restrict__ x,
                                                  const float* __restrict__ desc,
                                                  const float* __restrict__ w7,
                                                  const float* __restrict__ w3,
                                                  const float* __restrict__ alpha,
                                                  float* __restrict__ out) {
  __shared__ float sdesc[2 * 8 * WW];                 // 2ch x (2 rows + 6 halo) x 160
  __shared__ float swt[116];                          // 98 w7 + 18 w3
  __shared__ __align__(16) float wtile[2 * WW];       // sigmoid weights for the strip

  const int tid = threadIdx.x;
  const int b   = blockIdx.x / 80;
  const int h0  = (blockIdx.x - b * 80) * 2;

  if (tid < 116) swt[tid] = (tid < 98) ? w7[tid] : w3[tid - 98];

  // stage desc halo tile (zero-padded rows outside the image)
  for (int i = tid; i < 2 * 8 * WW; i += 256) {
    int ch   = i / (8 * WW);
    int rem  = i - ch * (8 * WW);
    int rr   = rem / WW;
    int w    = rem - rr * WW;
    int grow = h0 + rr - 3;
    float v = 0.0f;
    if (grow >= 0 && grow < HH)
      v = desc[(size_t)(b * 2 + ch) * HWSZ + grow * WW + w];
    sdesc[i] = v;
  }
  __syncthreads();

  const float al = alpha[0];

  // per-pixel fused conv + sigmoid (320 pixels per strip)
  for (int q = tid; q < 2 * WW; q += 256) {
    int y  = q / WW;
    int xw = q - y * WW;
    float acc7 = 0.0f, acc3 = 0.0f;
#pragma unroll
    for (int c = 0; c < 2; ++c) {
      const float* sp = sdesc + c * (8 * WW);
#pragma unroll
      for (int dy = 0; dy < 7; ++dy) {
        const float* row = sp + (y + dy) * WW;       // tile row 0 == image row h0-3
#pragma unroll
        for (int dx = 0; dx < 7; ++dx) {
          int xi = xw + dx - 3;
          if (xi >= 0 && xi < WW)
            acc7 += row[xi] * swt[c * 49 + dy * 7 + dx];
        }
      }
#pragma unroll
      for (int dy = 0; dy < 3; ++dy) {
        const float* row = sp + (y + dy + 2) * WW;   // 3x3, pad 1
#pragma unroll
        for (int dx = 0; dx < 3; ++dx) {
          int xi = xw + dx - 1;
          if (xi >= 0 && xi < WW)
            acc3 += row[xi] * swt[98 + c * 9 + dy * 3 + dx];
        }
      }
    }
    float v = al * acc7 + (1.0f - al) * acc3;
    wtile[q] = 1.0f / (1.0f + __expf(-v));
  }
  __syncthreads();

  // streaming multiply: 256 channels x 80 float4 per strip, coalesced
  const size_t xoff = (size_t)b * CH * HWSZ + (size_t)h0 * WW;
  const float4* wt4 = (const float4*)wtile;
  for (int u = tid; u < CH * 80; u += 256) {
    int c = u / 80;
    int i = u - c * 80;
    size_t off = xoff + (size_t)c * HWSZ;
    float4 xv = ((const float4*)(x + off))[i];
    float4 wv = wt4[i];
    float4 ov = make_float4(xv.x * wv.x, xv.y * wv.y, xv.z * wv.z, xv.w * wv.w);
    ((float4*)(out + off))[i] = ov;
  }
}

extern "C" void kernel_launch(void* const* d_in, const int* in_sizes, int n_in,
                              void* d_out, int out_size, void* d_ws, size_t ws_size,
                              hipStream_t stream) {
  const float* x     = (const float*)d_in[0];
  const float* w7    = (const float*)d_in[1];  // [1,2,7,7] = 98 floats
  const float* w3    = (const float*)d_in[2];  // [1,2,3,3] = 18 floats
  const float* alpha = (const float*)d_in[3];  // scalar
  float* out  = (float*)d_out;
  float* desc = (float*)d_ws;                  // [8,2,160,160] = 1.6 MB scratch

  // 12800 waves, 8 per block
  k_reduce<<<(BATCH * GROUPS_PER_B) / 8, 256, 0, stream>>>(x, desc);
  // 8 batches x 80 two-row strips
  k_conv_mul<<<BATCH * 80, 256, 0, stream>>>(x, desc, w7, w3, alpha, out);
}